// LightGCN_5995774345235
// MI455X (gfx1250) — compile-verified
//
#include <hip/hip_runtime.h>
#include <hip/hip_bf16.h>

#define NUM_USERS 100000
#define NUM_ITEMS 200000
#define EMB_DIM   64
#define N_NODES   (NUM_USERS + NUM_ITEMS)   // 300000
#define NUM_EDGES 9600000
#define EPB       64                        // edges staged per block

typedef float v2f __attribute__((ext_vector_type(2)));
typedef float v8f __attribute__((ext_vector_type(8)));

// ---------------------------------------------------------------------------
// embs0 = concat(user_emb, item_emb)
// ---------------------------------------------------------------------------
__global__ void lgcn_concat(const float* __restrict__ u,
                            const float* __restrict__ it,
                            float* __restrict__ out) {
    size_t i  = (size_t)blockIdx.x * blockDim.x + threadIdx.x;
    size_t n4 = (size_t)N_NODES * EMB_DIM / 4;
    if (i >= n4) return;
    size_t nu4 = (size_t)NUM_USERS * EMB_DIM / 4;
    float4 v = (i < nu4) ? ((const float4*)u)[i] : ((const float4*)it)[i - nu4];
    ((float4*)out)[i] = v;
}

__global__ void lgcn_zero(float4* __restrict__ p, size_t n4) {
    size_t i = (size_t)blockIdx.x * blockDim.x + threadIdx.x;
    if (i < n4) p[i] = make_float4(0.f, 0.f, 0.f, 0.f);
}

// ---------------------------------------------------------------------------
// CSR build step 1: row-degree histogram
// ---------------------------------------------------------------------------
__global__ void lgcn_hist(const int* __restrict__ row, int* __restrict__ deg, int n) {
    int i = blockIdx.x * blockDim.x + threadIdx.x;
    if (i < n) atomicAdd(&deg[row[i]], 1);
}

// ---------------------------------------------------------------------------
// CSR build step 2: exclusive prefix sum over 300k degrees (3 kernels)
// K1: per-block (1024 elems, 256 thr x 4) exclusive scan + block total
// ---------------------------------------------------------------------------
__global__ void __launch_bounds__(256) lgcn_scan_partial(
        const int* __restrict__ deg, int* __restrict__ out,
        int* __restrict__ blk_sums, int n) {
    __shared__ int s[256];
    const int tid  = threadIdx.x;
    const int base = blockIdx.x * 1024 + tid * 4;
    int v0 = (base + 0 < n) ? deg[base + 0] : 0;
    int v1 = (base + 1 < n) ? deg[base + 1] : 0;
    int v2 = (base + 2 < n) ? deg[base + 2] : 0;
    int v3 = (base + 3 < n) ? deg[base + 3] : 0;
    const int tsum = v0 + v1 + v2 + v3;
    s[tid] = tsum;
    __syncthreads();
    for (int off = 1; off < 256; off <<= 1) {          // Hillis-Steele inclusive
        int t = (tid >= off) ? s[tid - off] : 0;
        __syncthreads();
        s[tid] += t;
        __syncthreads();
    }
    int excl = s[tid] - tsum;
    if (base + 0 < n) out[base + 0] = excl;
    if (base + 1 < n) out[base + 1] = excl + v0;
    if (base + 2 < n) out[base + 2] = excl + v0 + v1;
    if (base + 3 < n) out[base + 3] = excl + v0 + v1 + v2;
    if (tid == 255) blk_sums[blockIdx.x] = s[255];
}

// K2: single block scans the (<=512) block totals -> exclusive
__global__ void __launch_bounds__(512) lgcn_scan_tops(int* __restrict__ blk_sums, int nblk) {
    __shared__ int s[512];
    const int tid = threadIdx.x;
    int v = (tid < nblk) ? blk_sums[tid] : 0;
    s[tid] = v;
    __syncthreads();
    for (int off = 1; off < 512; off <<= 1) {
        int t = (tid >= off) ? s[tid - off] : 0;
        __syncthreads();
        s[tid] += t;
        __syncthreads();
    }
    if (tid < nblk) blk_sums[tid] = s[tid] - v;
}

// K3: add block offsets; init scatter cursors; close row_ptr
__global__ void lgcn_scan_finalize(int* __restrict__ row_start, int* __restrict__ cursor,
                                   const int* __restrict__ blk_sums, int n) {
    int i = blockIdx.x * blockDim.x + threadIdx.x;
    if (i < n) {
        int v = row_start[i] + blk_sums[i >> 10];
        row_start[i] = v;
        cursor[i]    = v;
    }
    if (i == 0) row_start[n] = NUM_EDGES;
}

// ---------------------------------------------------------------------------
// CSR build step 3: scatter edges into row order, packing (col, val) as int2.
// Edge triples staged through the CDNA5 async global->LDS DMA path (ASYNCcnt).
// Block = 192 threads: all stage (one dword each), first 64 place edges.
// ---------------------------------------------------------------------------
__global__ void __launch_bounds__(192) lgcn_scatter(
        const int*   __restrict__ row,
        const int*   __restrict__ col,
        const float* __restrict__ val,
        int*  __restrict__ cursor,
        int2* __restrict__ edge_pack, int n_edges) {
    __shared__ int   s_row[EPB];
    __shared__ int   s_col[EPB];
    __shared__ float s_val[EPB];
    const int tid  = threadIdx.x;
    const int base = blockIdx.x * EPB;

    {   // async stage: 192 lanes, one dword each (row | col | val)
        const int arr = tid >> 6;
        const int idx = tid & (EPB - 1);
        const int g   = base + idx;
        if (g < n_edges) {
            const void* gsrc;
            const void* ldst;
            if      (arr == 0) { gsrc = (const void*)(row + g); ldst = (const void*)&s_row[idx]; }
            else if (arr == 1) { gsrc = (const void*)(col + g); ldst = (const void*)&s_col[idx]; }
            else               { gsrc = (const void*)(val + g); ldst = (const void*)&s_val[idx]; }
            unsigned lds_addr = (unsigned)(uintptr_t)ldst;   // low 32b of flat LDS ptr
            asm volatile("global_load_async_to_lds_b32 %0, %1, off"
                         :: "v"(lds_addr), "v"(gsrc) : "memory");
        }
    }
    asm volatile("s_wait_asynccnt 0" ::: "memory");
    __syncthreads();

    if (tid < EPB) {
        const int g = base + tid;
        if (g < n_edges) {
            const int r = s_row[tid];
            const int p = atomicAdd(&cursor[r], 1);
            edge_pack[p] = make_int2(s_col[tid], __float_as_int(s_val[tid]));
        }
    }
}

// ---------------------------------------------------------------------------
// CSR SpMM layer (no atomics, no zero pass): out[r] = sum val*x[col].
// 16 lanes per row, float4 per lane; gathers are full 256B segments and both
// ping-pong tables (2 x 76.8MB) live in the 192MB L2.
// ---------------------------------------------------------------------------
__global__ void __launch_bounds__(256) lgcn_spmm_csr(
        const int*  __restrict__ row_start,
        const int2* __restrict__ ep,
        const float* __restrict__ xin,
        float*       __restrict__ xout) {
    const int r  = (blockIdx.x * 256 + threadIdx.x) >> 4;
    const int d  = (threadIdx.x & 15) * 4;
    if (r >= N_NODES) return;
    const int s = row_start[r];
    const int e = row_start[r + 1];
    float ax = 0.f, ay = 0.f, az = 0.f, aw = 0.f;
    for (int j = s; j < e; ++j) {
        const int2  cv = ep[j];
        const float v  = __int_as_float(cv.y);
        const float4 xv = *(const float4*)(xin + (size_t)cv.x * EMB_DIM + d);
        ax += v * xv.x; ay += v * xv.y; az += v * xv.z; aw += v * xv.w;
    }
    *(float4*)(xout + (size_t)r * EMB_DIM + d) = make_float4(ax, ay, az, aw);
}

// ---------------------------------------------------------------------------
// Fallback SpMM (atomic scatter) when workspace is too small for CSR.
// Same async-LDS edge staging as round 1.
// ---------------------------------------------------------------------------
__global__ void __launch_bounds__(256) lgcn_spmm_atomic(
        const int*   __restrict__ row,
        const int*   __restrict__ col,
        const float* __restrict__ val,
        const float* __restrict__ xin,
        float*       __restrict__ xout, int n_edges) {
    __shared__ int   s_row[EPB];
    __shared__ int   s_col[EPB];
    __shared__ float s_val[EPB];
    const int tid  = threadIdx.x;
    const int base = blockIdx.x * EPB;

    if (tid < 3 * EPB) {
        const int arr = tid >> 6;
        const int idx = tid & (EPB - 1);
        const int g   = base + idx;
        if (g < n_edges) {
            const void* gsrc;
            const void* ldst;
            if      (arr == 0) { gsrc = (const void*)(row + g); ldst = (const void*)&s_row[idx]; }
            else if (arr == 1) { gsrc = (const void*)(col + g); ldst = (const void*)&s_col[idx]; }
            else               { gsrc = (const void*)(val + g); ldst = (const void*)&s_val[idx]; }
            unsigned lds_addr = (unsigned)(uintptr_t)ldst;
            asm volatile("global_load_async_to_lds_b32 %0, %1, off"
                         :: "v"(lds_addr), "v"(gsrc) : "memory");
        }
    }
    asm volatile("s_wait_asynccnt 0" ::: "memory");
    __syncthreads();

    const int el0 = tid >> 4;
    const int d   = (tid & 15) * 4;
#pragma unroll
    for (int p = 0; p < EPB / 16; ++p) {
        const int el = p * 16 + el0;
        const int g  = base + el;
        if (g < n_edges) {
            const int   r = s_row[el];
            const int   c = s_col[el];
            const float v = s_val[el];
            const float4 xv = *(const float4*)(xin + (size_t)c * EMB_DIM + d);
            float* dst = xout + (size_t)r * EMB_DIM + d;
            unsafeAtomicAdd(dst + 0, v * xv.x);
            unsafeAtomicAdd(dst + 1, v * xv.y);
            unsafeAtomicAdd(dst + 2, v * xv.z);
            unsafeAtomicAdd(dst + 3, v * xv.w);
        }
    }
}

// ---------------------------------------------------------------------------
// final = mean(embs0..3) as D = A x B with V_WMMA_F32_16X16X4_F32:
//   A = 0.25 * ones(16x4)  (constant -> layout-independent)
//   B(4x16) = 16 consecutive floats from each of the 4 embeddings (K slots
//             interchangeable: mean is symmetric in K)
//   D row M=0 (VGPR0, lanes 0..15, N=lane, per documented C/D layout) = result
// Exact f32 math; 4 WMMAs (64 outputs) per wave iteration.
// ---------------------------------------------------------------------------
#if __has_builtin(__builtin_amdgcn_wmma_f32_16x16x4_f32)
__global__ void __launch_bounds__(256) lgcn_mean_wmma(
        const float* __restrict__ e0, const float* __restrict__ e1,
        const float* __restrict__ e2, const float* __restrict__ e3,
        float* __restrict__ out, long long n) {
    const int  tid  = blockIdx.x * 256 + threadIdx.x;
    const int  wave = tid >> 5;                 // wave-uniform
    const int  lane = threadIdx.x & 31;
    const int  l    = lane & 15;
    const bool hi   = lane >= 16;
    const long long base = (long long)wave * 64;
    if (base >= n) return;                      // wave-uniform guard (EXEC all-1)

    const float* pa = hi ? e1 : e0;             // K-slot assignment arbitrary
    const float* pb = hi ? e3 : e2;
    v2f A; A[0] = 0.25f; A[1] = 0.25f;
#pragma unroll
    for (int c = 0; c < 4; ++c) {
        const long long off = base + c * 16 + l;
        v2f B; B[0] = pa[off]; B[1] = pb[off];
        v8f C = {};
        C = __builtin_amdgcn_wmma_f32_16x16x4_f32(
                /*neg_a=*/false, A, /*neg_b=*/false, B,
                /*c_mod=*/(short)0, C, /*reuse_a=*/false, /*reuse_b=*/false);
        if (!hi) out[off] = C[0];               // D[M=0, N=l]
    }
}
#define HAVE_WMMA_MEAN 1
#else
#define HAVE_WMMA_MEAN 0
#endif

__global__ void lgcn_mean_valu(const float4* __restrict__ e0, const float4* __restrict__ e1,
                               const float4* __restrict__ e2, const float4* __restrict__ e3,
                               float4* __restrict__ out, size_t n4) {
    size_t i = (size_t)blockIdx.x * blockDim.x + threadIdx.x;
    if (i >= n4) return;
    float4 a = e0[i], b = e1[i], c = e2[i], dd = e3[i];
    out[i] = make_float4(0.25f * (a.x + b.x + c.x + dd.x),
                         0.25f * (a.y + b.y + c.y + dd.y),
                         0.25f * (a.z + b.z + c.z + dd.z),
                         0.25f * (a.w + b.w + c.w + dd.w));
}

extern "C" void kernel_launch(void* const* d_in, const int* in_sizes, int n_in,
                              void* d_out, int out_size, void* d_ws, size_t ws_size,
                              hipStream_t stream) {
    (void)in_sizes; (void)n_in; (void)out_size;

    const float* user_emb = (const float*)d_in[0];
    const float* item_emb = (const float*)d_in[1];
    const int*   adj_row  = (const int*)d_in[2];
    const int*   adj_col  = (const int*)d_in[3];
    const float* adj_val  = (const float*)d_in[4];
    float* out = (float*)d_out;

    const size_t NN = (size_t)N_NODES * EMB_DIM;   // 19.2M floats per embedding
    float* final_out = out;
    float* e0 = out + 1 * NN;
    float* e1 = out + 2 * NN;
    float* e2 = out + 3 * NN;
    float* e3 = out + 4 * NN;
    float* embs[4] = { e0, e1, e2, e3 };

    const size_t n4   = NN / 4;
    const int    ewgs = (int)((n4 + 255) / 256);

    // workspace layout for CSR path
    const int ROW_PAD = 300032;                     // N_NODES padded, /4 aligned
    int*  deg       = (int*)d_ws;                   // [ROW_PAD]
    int*  row_start = deg + ROW_PAD;                // [ROW_PAD] (+1 used)
    int*  cursor    = row_start + ROW_PAD;          // [ROW_PAD]
    int*  blk_sums  = cursor + ROW_PAD;             // [1024]
    int2* edge_pack = (int2*)(blk_sums + 1024);     // [NUM_EDGES]
    const size_t ws_need = (size_t)(3 * ROW_PAD + 1024) * 4 + (size_t)NUM_EDGES * 8;

    lgcn_concat<<<ewgs, 256, 0, stream>>>(user_emb, item_emb, e0);

    const int edge_blocks = (NUM_EDGES + EPB - 1) / EPB;    // 150000

    if (ws_size >= ws_need) {
        // ---- build CSR once; 3 gather-only SpMM layers (no atomics) ----
        const int nscan = (N_NODES + 1023) / 1024;          // 293 blocks
        lgcn_zero<<<(ROW_PAD / 4 + 255) / 256, 256, 0, stream>>>((float4*)deg, ROW_PAD / 4);
        lgcn_hist<<<(NUM_EDGES + 255) / 256, 256, 0, stream>>>(adj_row, deg, NUM_EDGES);
        lgcn_scan_partial<<<nscan, 256, 0, stream>>>(deg, row_start, blk_sums, N_NODES);
        lgcn_scan_tops<<<1, 512, 0, stream>>>(blk_sums, nscan);
        lgcn_scan_finalize<<<(N_NODES + 255) / 256, 256, 0, stream>>>(row_start, cursor,
                                                                      blk_sums, N_NODES);
        lgcn_scatter<<<edge_blocks, 192, 0, stream>>>(adj_row, adj_col, adj_val,
                                                      cursor, edge_pack, NUM_EDGES);
        const int spmm_blocks = (N_NODES * 16 + 255) / 256; // 18750
        for (int l = 0; l < 3; ++l)
            lgcn_spmm_csr<<<spmm_blocks, 256, 0, stream>>>(row_start, edge_pack,
                                                           embs[l], embs[l + 1]);
    } else {
        // ---- fallback: async-staged atomic scatter SpMM ----
        for (int l = 0; l < 3; ++l) {
            lgcn_zero<<<ewgs, 256, 0, stream>>>((float4*)embs[l + 1], n4);
            lgcn_spmm_atomic<<<edge_blocks, 256, 0, stream>>>(adj_row, adj_col, adj_val,
                                                              embs[l], embs[l + 1], NUM_EDGES);
        }
    }

#if HAVE_WMMA_MEAN
    {   // 64 floats per wave; NN divisible by 64 -> exact grid
        const long long nwaves = (long long)(NN / 64);              // 300000
        const int blocks = (int)((nwaves * 32 + 255) / 256);        // 37500
        lgcn_mean_wmma<<<blocks, 256, 0, stream>>>(e0, e1, e2, e3, final_out, (long long)NN);
    }
#else
    lgcn_mean_valu<<<ewgs, 256, 0, stream>>>((const float4*)e0, (const float4*)e1,
                                             (const float4*)e2, (const float4*)e3,
                                             (float4*)final_out, n4);
#endif
}